// TreeCRFLayer_55241869361598
// MI455X (gfx1250) — compile-verified
//
#include <hip/hip_runtime.h>
#include <stdint.h>

#define LBL 2048       // N_LABELS
#define NLEV 7         // tree depth levels 0..6

typedef float v4f __attribute__((ext_vector_type(4)));  // native vector for NT stores

// Numerically-stable 2-way log-sum-exp using fast HW transcendentals.
__device__ __forceinline__ float lse2(float a, float b) {
    float m = fmaxf(a, b);
    float n = fminf(a, b);
    return m + __logf(1.0f + __expf(n - m));
}

__launch_bounds__(256, 1)
__global__ void treecrf_bp_kernel(const float* __restrict__ emissions, // (B,2,L)
                                  const float* __restrict__ pairs,     // (L,L,2,2)
                                  float* __restrict__ out)             // (B,2,L)
{
    // Level start indices for parent(i) = (i-1)/4, BRANCH=4, L=2048.
    constexpr int LS[NLEV + 1] = {0, 1, 5, 21, 85, 341, 1365, 2048};
    // Last node is 2047 -> last parent with any child is (2047-1)/4 = 511.
    constexpr int PMAX = 512;

    __shared__ __align__(16) float em[2 * LBL];   // emissions slice (16 KB)
    __shared__ __align__(16) float Au[2 * LBL];   // alpha accumulators
    __shared__ __align__(16) float Bd[2 * LBL];   // beta accumulators

    const int tid = threadIdx.x;
    const int b   = blockIdx.x;
    const float4* __restrict__ p4 = reinterpret_cast<const float4*>(pairs);

    // ---- CDNA5 async bulk copy: emissions[b] (16 KB) -> LDS, b128 per lane.
    // LDS byte address = low 32 bits of the flat pointer (ISA 10.2 aperture rule).
    {
        const unsigned long long gbase =
            (unsigned long long)(uintptr_t)(emissions + (size_t)b * (2 * LBL));
        const uint32_t lbase = (uint32_t)(uintptr_t)(&em[0]);
#pragma unroll
        for (int it = 0; it < 4; ++it) {                 // 4 * 256 lanes * 16B = 16 KB
            uint32_t voff  = (uint32_t)(it * 256 + tid) * 16u;
            uint32_t laddr = lbase + voff;
            asm volatile("global_load_async_to_lds_b128 %0, %1, %2"
                         :
                         : "v"(laddr), "v"(voff), "s"(gbase)
                         : "memory");
        }
    }

    // Zero accumulators while the async copies are in flight.
    // Exact trip count + b128 LDS stores (no strided-loop exec guards).
    {
        const v4f z4 = {0.0f, 0.0f, 0.0f, 0.0f};
        v4f* __restrict__ A4 = reinterpret_cast<v4f*>(Au);
        v4f* __restrict__ B4 = reinterpret_cast<v4f*>(Bd);
#pragma unroll
        for (int it = 0; it < 4; ++it) {                 // 4 * 256 * 4 floats = 4096
            A4[it * 256 + tid] = z4;
            B4[it * 256 + tid] = z4;
        }
    }

#if __has_builtin(__builtin_amdgcn_s_wait_asynccnt)
    __builtin_amdgcn_s_wait_asynccnt(0);
#else
    asm volatile("s_wait_asynccnt 0" ::: "memory");
#endif
    __syncthreads();

    // ---- Upward sweep (alphas), parent-centric gather (children of p are 4p+1..4p+4,
    // so the reference's scatter-add becomes a conflict-free gather; no LDS atomics).
    for (int dl = NLEV - 2; dl >= 0; --dl) {             // parent levels 5..0
        const int pend = (LS[dl + 1] < PMAX) ? LS[dl + 1] : PMAX;  // skip childless tail
        for (int p = LS[dl] + tid; p < pend; p += 256) {
            float a0 = 0.0f, a1 = 0.0f;
            const int c0 = 4 * p + 1;
#pragma unroll
            for (int j = 0; j < 4; ++j) {
                const int ch = c0 + j;
                if (ch < LBL) {                          // last level is partial
                    const float  l0 = em[ch]        + Au[ch];
                    const float  l1 = em[LBL + ch]  + Au[LBL + ch];
                    const float4 t  = p4[(size_t)p * LBL + ch]; // pairs[p][ch][yp][yc]
                    a0 += lse2(t.x + l0, t.y + l1);      // yp = 0
                    a1 += lse2(t.z + l0, t.w + l1);      // yp = 1
                }
            }
            Au[p]       += a0;
            Au[LBL + p] += a1;
        }
        __syncthreads();
    }

    // ---- Downward sweep (betas), each child gathers from its (finished) parent.
    for (int d = 1; d < NLEV; ++d) {
        for (int k = LS[d] + tid; k < LS[d + 1]; k += 256) {
            const int   p  = (k - 1) >> 2;
            const float l0 = em[p]       + Bd[p];
            const float l1 = em[LBL + p] + Bd[LBL + p];
            const float4 t = p4[(size_t)k * LBL + p];    // pairs[k][p][yc][yp]
            Bd[k]       = lse2(t.x + l0, t.y + l1);      // yc = 0
            Bd[LBL + k] = lse2(t.z + l0, t.w + l1);      // yc = 1
        }
        __syncthreads();
    }

    // ---- Normalize over the 2 classes; stream out with non-temporal b128 stores
    // (output is write-once: keep L2 free for the shared `pairs` edge entries).
    v4f* __restrict__ o4 = reinterpret_cast<v4f*>(out + (size_t)b * (2 * LBL));
#pragma unroll
    for (int it = 0; it < 2; ++it) {                     // 2 * 256 * 4 = 2048 labels
        const int q = it * 256 + tid;
        const int i = q * 4;
        float s0[4], s1[4];
#pragma unroll
        for (int j = 0; j < 4; ++j) {
            s0[j] = em[i + j]       + Au[i + j]       + Bd[i + j];
            s1[j] = em[LBL + i + j] + Au[LBL + i + j] + Bd[LBL + i + j];
            const float z = lse2(s0[j], s1[j]);
            s0[j] -= z;
            s1[j] -= z;
        }
        const v4f r0 = {s0[0], s0[1], s0[2], s0[3]};
        const v4f r1 = {s1[0], s1[1], s1[2], s1[3]};
        __builtin_nontemporal_store(r0, &o4[q]);               // class 0 row
        __builtin_nontemporal_store(r1, &o4[LBL / 4 + q]);     // class 1 row
    }
}

extern "C" void kernel_launch(void* const* d_in, const int* in_sizes, int n_in,
                              void* d_out, int out_size, void* d_ws, size_t ws_size,
                              hipStream_t stream) {
    const float* emissions = (const float*)d_in[0];   // (B, 2, 2048) f32
    const float* pairs     = (const float*)d_in[1];   // (2048, 2048, 2, 2) f32
    float*       out       = (float*)d_out;           // (B, 2, 2048) f32

    const int batch = in_sizes[0] / (2 * LBL);        // = 2048
    treecrf_bp_kernel<<<batch, 256, 0, stream>>>(emissions, pairs, out);
    (void)d_ws; (void)ws_size; (void)n_in; (void)out_size;
}